// Causal_ALIFLayer2D_52544629899446
// MI455X (gfx1250) — compile-verified
//
#include <hip/hip_runtime.h>
#include <math.h>

// Problem constants (from reference): B=16, T=64, H=32, W=32, C=64
#define BB    16
#define TT    64
#define HWC   65536            // 32*32*64
#define NTOT  (BB * TT * HWC)  // 67,108,864 elements per output tensor
#define DEPTH 4                // async prefetch pipeline depth (timesteps in flight)
#define BLK   256

typedef __attribute__((ext_vector_type(4))) float v4f;

__device__ __forceinline__ float softplus_f(float x) {
    // matches jax.nn.softplus to fp32 accuracy for the parameter magnitudes here
    return (x > 20.0f) ? x : log1pf(expf(x));
}
__device__ __forceinline__ float sigmoid_f(float x) {
    return 1.0f / (1.0f + expf(-x));
}

__global__ __launch_bounds__(BLK)
void alif_scan_kernel(const float* __restrict__ x,
                      const float* __restrict__ hp_base_step,
                      const float* __restrict__ hp_base_decay,
                      const float* __restrict__ step_w_raw,
                      const float* __restrict__ decay_w_raw,
                      const float* __restrict__ gamma,
                      const float* __restrict__ beta,
                      float* __restrict__ out)
{
    // Per-lane staging slots: DEPTH timesteps x 256 lanes x 16B = 16 KB LDS.
    __shared__ v4f sbuf[DEPTH * BLK];

    const int tid = threadIdx.x;
    const int g   = blockIdx.x * BLK + tid;        // float4 unit id, 0..262143
    const int n0  = g << 2;                        // first of this thread's 4 channels
    const int b   = n0 >> 16;                      // n0 / HWC
    const int r   = n0 & (HWC - 1);                // flat (h,w,c) index, multiple of 4

    // ---- loop-invariant per-neuron effective parameters (computed once) ----
    const float base_step  = hp_base_step[0];
    const float base_decay = hp_base_decay[0];

    const v4f g4 = *(const v4f*)(gamma       + r);
    const v4f b4 = *(const v4f*)(beta        + r);
    const v4f sw = *(const v4f*)(step_w_raw  + r);
    const v4f dw = *(const v4f*)(decay_w_raw + r);

    v4f step_eff, decay_eff;
    step_eff.x  = base_step * softplus_f(sw.x);
    step_eff.y  = base_step * softplus_f(sw.y);
    step_eff.z  = base_step * softplus_f(sw.z);
    step_eff.w  = base_step * softplus_f(sw.w);
    decay_eff.x = base_decay + (1.0f - base_decay) * sigmoid_f(dw.x);
    decay_eff.y = base_decay + (1.0f - base_decay) * sigmoid_f(dw.y);
    decay_eff.z = base_decay + (1.0f - base_decay) * sigmoid_f(dw.z);
    decay_eff.w = base_decay + (1.0f - base_decay) * sigmoid_f(dw.w);

    // ---- global addressing: timestep stride = HWC floats = 16384 float4 ----
    const int     tstep = HWC / 4;
    const size_t  base4 = (size_t)b * TT * tstep + (size_t)(r >> 2);
    const v4f* __restrict__ xp = (const v4f*)x + base4;      // x[b, t, r..r+3]
    v4f* __restrict__ sp = (v4f*)out + base4;                // spikes[b, t, ...]
    v4f* __restrict__ vp = sp + (size_t)(NTOT / 4);          // voltages[b, t, ...]

    // Raw workgroup-relative LDS byte offset of this lane's slot-0 element.
    // (generic-address low 32 bits == LDS offset on gfx1250; slot stride = 4 KB)
    const unsigned lds_base = (unsigned)(size_t)(&sbuf[tid]);

    // ---- prologue: fill the DEPTH-deep async pipeline (T=64 >= DEPTH) ----
#pragma unroll
    for (int d = 0; d < DEPTH; ++d) {
        const v4f* gp = xp + (size_t)d * tstep;
        const unsigned loff = lds_base + ((unsigned)d << 12);
        asm volatile("global_load_async_to_lds_b128 %0, %1, off"
                     :: "v"(loff), "v"(gp)
                     : "memory");
    }

    v4f v   = (v4f)0.0f;
    v4f vth = (v4f)0.0f;

#pragma unroll 4
    for (int t = 0; t < TT; ++t) {
        const int slot = t & (DEPTH - 1);

        // Wait until the async copy for timestep t has landed in LDS
        // (ASYNCcnt decrements in order; <= DEPTH-1 outstanding => t done).
        asm volatile("s_wait_asynccnt 3" ::: "memory");

        const v4f xv = sbuf[(slot << 8) + tid];              // ds_load_b128

        // v_t = v*0.8 + (x*gamma + beta)   (pre-reset membrane potential)
        v4f vme = v * 0.8f + (xv * g4 + b4);
        v4f vth_eff = vth + 0.5f;
        v4f d = vme - vth_eff;

        v4f s, vn;
        s.x = d.x > 0.0f ? 1.0f : 0.0f;  vn.x = d.x > 0.0f ? d.x : vme.x;
        s.y = d.y > 0.0f ? 1.0f : 0.0f;  vn.y = d.y > 0.0f ? d.y : vme.y;
        s.z = d.z > 0.0f ? 1.0f : 0.0f;  vn.z = d.z > 0.0f ? d.z : vme.z;
        s.w = d.w > 0.0f ? 1.0f : 0.0f;  vn.w = d.w > 0.0f ? d.w : vme.w;

        // write-once output streams: bypass-ish via non-temporal hint
        __builtin_nontemporal_store(s,   sp + (size_t)t * tstep);
        __builtin_nontemporal_store(vme, vp + (size_t)t * tstep);

        v   = vn;
        vth = vth * decay_eff + s * step_eff;

        // Refill this slot with timestep t+DEPTH (clamped tail re-reads t=T-1;
        // its slot is never consumed again, so the data is irrelevant).
        int tn = t + DEPTH;
        tn = (tn < TT) ? tn : (TT - 1);
        const v4f* gp = xp + (size_t)tn * tstep;
        const unsigned loff = lds_base + ((unsigned)slot << 12);
        // s_wait_dscnt 0 first: the DS read above must drain before the async
        // engine (unordered vs the DS pipe) overwrites the same LDS slot.
        asm volatile("s_wait_dscnt 0\n\t"
                     "global_load_async_to_lds_b128 %0, %1, off"
                     :: "v"(loff), "v"(gp)
                     : "memory");
    }
}

extern "C" void kernel_launch(void* const* d_in, const int* in_sizes, int n_in,
                              void* d_out, int out_size, void* d_ws, size_t ws_size,
                              hipStream_t stream) {
    // setup_inputs() order:
    // 0: x [B,T,H,W,C] f32      1: hp_alpha (unused in fwd)
    // 2: hp_base_step           3: hp_base_decay
    // 4: step_w_raw [H,W,C]     5: decay_w_raw [H,W,C]
    // 6: gamma [H,W,C]          7: beta [H,W,C]
    const float* x   = (const float*)d_in[0];
    const float* bs  = (const float*)d_in[2];
    const float* bd  = (const float*)d_in[3];
    const float* swr = (const float*)d_in[4];
    const float* dwr = (const float*)d_in[5];
    const float* gm  = (const float*)d_in[6];
    const float* bt  = (const float*)d_in[7];
    float* out = (float*)d_out;   // [spikes(B,T,H,W,C) | voltages(B,T,H,W,C)]

    const int n4 = NTOT / TT / 4 * TT / TT;  // neurons/4 per... (kept explicit below)
    (void)n4; (void)in_sizes; (void)n_in; (void)out_size; (void)d_ws; (void)ws_size;

    // B*H*W*C/4 = 262144 float4 lanes -> 1024 blocks of 256 threads
    dim3 grid(1024), block(BLK);
    alif_scan_kernel<<<grid, block, 0, stream>>>(x, bs, bd, swr, dwr, gm, bt, out);
}